// ShiftedPosBias_23845658427614
// MI455X (gfx1250) — compile-verified
//
#include <hip/hip_runtime.h>
#include <cstdint>

// ShiftedPosBias: out[h1][w1][h2][w2] = biases[h2-h1+R][w2-w1+R] if in range else 0
// H = W = 100, R = 8, bias table 17x17 f32. Output 100^4 f32 = 400 MB.
// Pure store-bandwidth kernel (~17 us at 23.3 TB/s on MI455X HBM).
// Bias table staged into LDS once per workgroup via the gfx1250 Tensor Data
// Mover (TENSORcnt), read back with explicit ds_load_b32 (DScnt), output
// streamed with non-temporal 128-bit stores.

#define RR  8
#define HH  100
#define WW  100
#define BW  (2 * RR + 1)   // 17

typedef __attribute__((ext_vector_type(4))) unsigned int u32x4;
typedef __attribute__((ext_vector_type(4))) int          i32x4;
typedef __attribute__((ext_vector_type(8))) int          i32x8;
typedef __attribute__((ext_vector_type(4))) float        f32x4;

__global__ __launch_bounds__(256) void
ShiftedPosBias_23845658427614_kernel(const float* __restrict__ biases,
                                     float* __restrict__ out,
                                     int n4)  // number of float4 elements (out_size/4)
{
    __shared__ float s_bias[BW * BW];

    // LDS byte offset of s_bias: low 32 bits of the generic pointer to an LDS
    // object are the ds-relative byte offset. This also escapes s_bias so the
    // allocation (and the TDM's target) cannot be optimized away.
    uint32_t lds_base = (uint32_t)(uintptr_t)(void*)&s_bias[0];

    // ---- Stage 17x17 bias table into LDS with the Tensor Data Mover ----
    // Wave 0 issues the DMA and drains TENSORcnt; the workgroup barrier then
    // releases the remaining 7 waves.
    if (threadIdx.x < 32) {
        uint64_t ga = (uint64_t)(uintptr_t)biases;

        // D# group 0 (128b): count=1, lds_addr, global_addr, type=2
        u32x4 g0;
        g0[0] = 1u;                                   // count=1 (valid), user mode
        g0[1] = lds_base;                             // lds_addr (bytes)
        g0[2] = (uint32_t)(ga & 0xFFFFFFFFu);         // global_addr[31:0]
        g0[3] = (uint32_t)((ga >> 32) & 0x01FFFFFFu)  // global_addr[56:32]
              | (2u << 30);                           // type = 2 ("image")

        // D# group 1 (256b): data_size=4B, tensor 17x17, tile 17x17, stride0=17
        i32x8 g1;
        g1[0] = (2 << 16);                 // workgroup_mask=0 | data_size=2 (4 bytes)
        g1[1] = (BW << 16);                // atomic_barrier_addr=0 | tensor_dim0 lo16=17
        g1[2] = (BW << 16);                // tensor_dim0 hi16=0 | tensor_dim1 lo16=17
        g1[3] = (BW << 16);                // tensor_dim1 hi16=0 | tile_dim0=17
        g1[4] = BW;                        // tile_dim1=17 | tile_dim2=0
        g1[5] = BW;                        // tensor_dim0_stride lo32=17
        g1[6] = 0;                         // stride0 hi16=0 | tensor_dim1_stride lo16=0
        g1[7] = 0;

        i32x4 z4 = {0, 0, 0, 0};
#if __has_include(<hip/amd_detail/amd_gfx1250_TDM.h>)
        // therock / clang-23 toolchain: 6-arg builtin
        i32x8 z8 = {0, 0, 0, 0, 0, 0, 0, 0};
        __builtin_amdgcn_tensor_load_to_lds(g0, g1, z4, z4, z8, 0);
#else
        // ROCm 7.2 / clang-22 toolchain: 5-arg builtin
        __builtin_amdgcn_tensor_load_to_lds(g0, g1, z4, z4, 0);
#endif
        __builtin_amdgcn_s_wait_tensorcnt(0);
    }
    __syncthreads();

    // ---- Stream the 400 MB output, one float4 per thread ----
    int t = blockIdx.x * blockDim.x + threadIdx.x;
    if (t >= n4) return;
    int i  = t * 4;            // flat element index, < 1e8 fits in int32
    int w2 = i % WW;           // 4 consecutive w2 stay in one row (100 % 4 == 0)
    int r  = i / WW;
    int h2 = r % HH;
    int r2 = r / HH;
    int w1 = r2 % WW;
    int h1 = r2 / WW;

    int  dh = h2 - h1 + RR;
    bool vh = (unsigned)dh <= (unsigned)(2 * RR);
    int  dhc = vh ? dh : 0;                       // clamp: stay inside the table
    uint32_t rowoff = lds_base + (uint32_t)(dhc * BW) * 4u;

    int dwb = w2 - w1 + RR;                       // dw for k = 0
    bool vw0 = (unsigned)(dwb + 0) <= (unsigned)(2 * RR);
    bool vw1 = (unsigned)(dwb + 1) <= (unsigned)(2 * RR);
    bool vw2 = (unsigned)(dwb + 2) <= (unsigned)(2 * RR);
    bool vw3 = (unsigned)(dwb + 3) <= (unsigned)(2 * RR);
    uint32_t o0 = rowoff + 4u * (uint32_t)(vw0 ? dwb + 0 : 0);
    uint32_t o1 = rowoff + 4u * (uint32_t)(vw1 ? dwb + 1 : 0);
    uint32_t o2 = rowoff + 4u * (uint32_t)(vw2 ? dwb + 2 : 0);
    uint32_t o3 = rowoff + 4u * (uint32_t)(vw3 ? dwb + 3 : 0);

    // Explicit LDS reads: asm volatile so the compiler cannot fold them to
    // undef (the only writer of s_bias is the TDM, which AA cannot see).
    float r0, r1, r2x, r3;
    asm volatile(
        "ds_load_b32 %0, %4\n\t"
        "ds_load_b32 %1, %5\n\t"
        "ds_load_b32 %2, %6\n\t"
        "ds_load_b32 %3, %7\n\t"
        "s_wait_dscnt 0x0"
        : "=&v"(r0), "=&v"(r1), "=&v"(r2x), "=&v"(r3)
        : "v"(o0), "v"(o1), "v"(o2), "v"(o3));

    f32x4 v;
    v[0] = (vh && vw0) ? r0  : 0.0f;
    v[1] = (vh && vw1) ? r1  : 0.0f;
    v[2] = (vh && vw2) ? r2x : 0.0f;
    v[3] = (vh && vw3) ? r3  : 0.0f;

    // Non-temporal 128-bit store: output (400 MB) exceeds L2 (192 MB).
    __builtin_nontemporal_store(v, (f32x4*)(out + i));
}

extern "C" void kernel_launch(void* const* d_in, const int* in_sizes, int n_in,
                              void* d_out, int out_size, void* d_ws, size_t ws_size,
                              hipStream_t stream) {
    (void)in_sizes; (void)n_in; (void)d_ws; (void)ws_size;
    const float* biases = (const float*)d_in[1];   // d_in[0] = feat (shape only)
    float*       out    = (float*)d_out;

    int n4     = out_size / 4;                     // 25,000,000 float4 stores
    int block  = 256;                              // 8 waves per block (wave32)
    int blocks = (n4 + block - 1) / block;

    ShiftedPosBias_23845658427614_kernel<<<blocks, block, 0, stream>>>(biases, out, n4);
}